// TargetSpecificStructuralLoss_9380208574579
// MI455X (gfx1250) — compile-verified
//
#include <hip/hip_runtime.h>

// ext-vector types (clang-native, safe with __builtin_nontemporal_load)
typedef __attribute__((ext_vector_type(2))) float f2;
typedef __attribute__((ext_vector_type(4))) float f4;
typedef __attribute__((ext_vector_type(8))) float f8;

__device__ __forceinline__ float relu_(float x) { return x > 0.f ? x : 0.f; }

// One group of 10 floats: bid = p[0..4], ask = p[5..9]
// ask_viol: relu(a_k - a_{k+1}), bid_viol: relu(b_{k+1} - b_k), cross: relu(b0 - a0)
__device__ __forceinline__ float group_viol(const float* p) {
    float s;
    s  = relu_(p[5] - p[6]);
    s += relu_(p[6] - p[7]);
    s += relu_(p[7] - p[8]);
    s += relu_(p[8] - p[9]);
    s += relu_(p[1] - p[0]);
    s += relu_(p[2] - p[1]);
    s += relu_(p[3] - p[2]);
    s += relu_(p[4] - p[3]);
    s += relu_(p[0] - p[5]);
    return s;
}

// Stage 1: stream 335 MB with NT b128 loads, one f32 partial per block.
__global__ __launch_bounds__(256)
void tssl_stream_reduce(const float* __restrict__ in,
                        float* __restrict__ partials,
                        long long nfloat) {
    const int tid      = blockIdx.x * blockDim.x + threadIdx.x;
    const int nthreads = gridDim.x * blockDim.x;

    // 160-byte chunks = 4 groups = 10 x float4 (16B aligned).
    const long long nchunk = nfloat / 40;
    float acc = 0.f;

    for (long long c = tid; c < nchunk; c += nthreads) {
        const f4* src = (const f4*)(in + c * 40);
        float p[40];
#pragma unroll
        for (int i = 0; i < 10; ++i) {
            f4 v = __builtin_nontemporal_load(src + i);   // global_load_b128, NT
            p[4*i+0] = v.x; p[4*i+1] = v.y; p[4*i+2] = v.z; p[4*i+3] = v.w;
        }
#pragma unroll
        for (int g = 0; g < 4; ++g) acc += group_viol(p + 10 * g);
    }

    // Tail groups (none for the reference shape; kept for generality). 8B aligned.
    const long long ngroup = nfloat / 10;
    for (long long g = nchunk * 4 + tid; g < ngroup; g += nthreads) {
        const f2* q = (const f2*)(in + g * 10);
        float p[10];
#pragma unroll
        for (int i = 0; i < 5; ++i) {
            f2 v = __builtin_nontemporal_load(q + i);
            p[2*i] = v.x; p[2*i+1] = v.y;
        }
        acc += group_viol(p);
    }

    // wave32 shuffle reduce
#pragma unroll
    for (int off = 16; off > 0; off >>= 1) acc += __shfl_xor(acc, off, 32);

    __shared__ float lds[8];                 // 256 threads / 32 lanes = 8 waves
    const int wave = threadIdx.x >> 5;
    if ((threadIdx.x & 31) == 0) lds[wave] = acc;
    __syncthreads();
    if (threadIdx.x == 0) {
        float b = 0.f;
#pragma unroll
        for (int w = 0; w < 8; ++w) b += lds[w];
        partials[blockIdx.x] = b;
    }
}

// Stage 2: one wave. Sum block partials, then reduce the 32 lane partials with
// chained v_wmma_f32_16x16x4_f32 ops (EXEC is all ones: 32 live lanes).
//
// WMMA #1: A[m][0]=p_m, A[m][2]=p_{m+16} (A vgpr1 = 0), B = ones
//          -> D[j][n] = s_j = p_j + p_{j+16}   (row-sum replicated over n)
// WMMA #2..#5 (accumulating): A = ones, B = (d[j], d[j+1])
//          -> acc[*][*] += s_j + s_{j+1} + s_{j+8} + s_{j+9}
// After 4 accumulations every element of acc = sum over all 32 lanes.
__global__ __launch_bounds__(32)
void tssl_final_reduce(const float* __restrict__ partials, int n,
                       float* __restrict__ out, float scale) {
    float p = 0.f;
    for (int i = threadIdx.x; i < n; i += 32) p += partials[i];

    f2 a;    a.x = p;   a.y = 0.f;
    f2 ones; ones.x = 1.f; ones.y = 1.f;
    f8 z = {};

    f8 d = __builtin_amdgcn_wmma_f32_16x16x4_f32(
        /*neg_a=*/false, a, /*neg_b=*/false, ones,
        /*c_mod=*/(short)0, z, /*reuse_a=*/false, /*reuse_b=*/false);

    f8 acc = {};
#pragma unroll
    for (int j = 0; j < 8; j += 2) {
        f2 b; b.x = d[j]; b.y = d[j + 1];
        acc = __builtin_amdgcn_wmma_f32_16x16x4_f32(
            false, ones, false, b, (short)0, acc, false, false);
    }

    if (threadIdx.x == 0) out[0] = acc[0] * scale;
}

extern "C" void kernel_launch(void* const* d_in, const int* in_sizes, int n_in,
                              void* d_out, int out_size, void* d_ws, size_t ws_size,
                              hipStream_t stream) {
    const float* in  = (const float*)d_in[0];
    float*       out = (float*)d_out;
    float*       ws  = (float*)d_ws;

    const long long nfloat = (long long)in_sizes[0];   // 4096*2048*10
    const int blocks  = 2048;                          // 8 KB of ws partials
    const int threads = 256;                           // 8 wave32 per block

    tssl_stream_reduce<<<blocks, threads, 0, stream>>>(in, ws, nfloat);
    // mean(axis=0) over 4096 rows -> grand_sum / 4096
    tssl_final_reduce<<<1, 32, 0, stream>>>(ws, blocks, out, 1.0f / 4096.0f);
}